// ImpCloudMMD_53025666236932
// MI455X (gfx1250) — compile-verified
//
#include <hip/hip_runtime.h>
#include <hip/hip_bf16.h>

typedef float v2f __attribute__((ext_vector_type(2)));
typedef float v8f __attribute__((ext_vector_type(8)));

#define KEXP (-0.7213475204444817f)  // -1/(2 ln2), sigma = 1

// Per-lane sum of exp(-d2/2) for one 16x16 Gram tile:
// A-tile operand/norms already resident; loads the B tile, runs one
// V_WMMA_F32_16X16X4_F32 (K=3 padded to 4) and the exp epilogue.
__device__ __forceinline__ float tile_pair_sum(
    v2f aop, const float* __restrict__ a2v,
    const float* __restrict__ B, int tb, int col, bool hi) {
  const float* bp = B + (size_t)(tb * 16 + col) * 3;
  float bx = bp[0], by = bp[1], bz = bp[2];
  float nb = bx * bx + by * by + bz * bz;
  // Operand layout (mirrors A): lanes 0-15 -> (x,y) = K0,K1 ; lanes 16-31 -> (z,0) = K2,K3
  v2f bop;
  bop.x = hi ? bz : bx;
  bop.y = hi ? 0.0f : by;

  v8f c = {};
  c = __builtin_amdgcn_wmma_f32_16x16x4_f32(
      /*neg_a=*/false, aop, /*neg_b=*/false, bop,
      /*c_mod=*/(short)0, c, /*reuse_a=*/false, /*reuse_b=*/false);

  float s = 0.0f;
#pragma unroll
  for (int r = 0; r < 8; ++r) {
    float d2 = a2v[r] + nb - 2.0f * c[r];     // ||a||^2 + ||b||^2 - 2 a.b
    d2 = fmaxf(d2, 0.0f);
    s += __builtin_amdgcn_exp2f(d2 * KEXP);   // exp(-d2/2) via v_exp_f32
  }
  return s;
}

// Load the A-tile operand + per-row norms (broadcast to C/D layout rows).
__device__ __forceinline__ void load_tile(
    const float* __restrict__ A, int ta, int col, bool hi,
    v2f* aop, float* a2v) {
  const float* ap = A + (size_t)(ta * 16 + col) * 3;
  float ax = ap[0], ay = ap[1], az = ap[2];
  float na = ax * ax + ay * ay + az * az;
  aop->x = hi ? az : ax;
  aop->y = hi ? 0.0f : ay;
  const int base = hi ? 8 : 0;  // C/D: VGPR r holds M = r + 8*(lane/16)
#pragma unroll
  for (int r = 0; r < 8; ++r) a2v[r] = __shfl(na, base + r, 32);
}

// ---------------- Full (non-symmetric) Gram: k(x, w) ----------------
__global__ __launch_bounds__(256) void mmd_gram_full_kernel(
    const float* __restrict__ A, const float* __restrict__ B,
    int nTiles, float coef, float* __restrict__ partial) {
  const int lane = threadIdx.x & 31;
  const int ta   = blockIdx.x * (blockDim.x >> 5) + (threadIdx.x >> 5);
  if (ta >= nTiles) return;  // wave-uniform: EXEC all-1s where active
  const int  col = lane & 15;
  const bool hi  = lane >= 16;

  v2f aop; float a2v[8];
  load_tile(A, ta, col, hi, &aop, a2v);

  float sum = 0.0f;
  for (int tb = 0; tb < nTiles; ++tb)
    sum += tile_pair_sum(aop, a2v, B, tb, col, hi);

#pragma unroll
  for (int off = 16; off > 0; off >>= 1) sum += __shfl_xor(sum, off, 32);
  if (lane == 0) partial[ta] = sum * coef;
}

// ---------------- Symmetric Gram: k(w,w) / k(x,x) ----------------
// Upper-triangular tiles only; off-diagonal tiles weighted 2.
// Wave s owns row-tiles {s, nTiles-1-s} -> exactly nTiles+1 iterations per wave.
__global__ __launch_bounds__(256) void mmd_gram_sym_kernel(
    const float* __restrict__ P, int nTiles, float coef,
    float* __restrict__ partial) {
  const int lane  = threadIdx.x & 31;
  const int s     = blockIdx.x * (blockDim.x >> 5) + (threadIdx.x >> 5);
  const int nHalf = nTiles >> 1;
  if (s >= nHalf) return;  // wave-uniform
  const int  col = lane & 15;
  const bool hi  = lane >= 16;

  float sumD = 0.0f;  // diagonal tiles (weight 1)
  float sumO = 0.0f;  // off-diagonal tiles (weight 2)

  const int rows[2] = {s, nTiles - 1 - s};
#pragma unroll
  for (int k = 0; k < 2; ++k) {
    const int ta = rows[k];
    v2f aop; float a2v[8];
    load_tile(P, ta, col, hi, &aop, a2v);
    sumD += tile_pair_sum(aop, a2v, P, ta, col, hi);       // tb == ta
    for (int tb = ta + 1; tb < nTiles; ++tb)               // tb > ta
      sumO += tile_pair_sum(aop, a2v, P, tb, col, hi);
  }

  float sum = sumD + 2.0f * sumO;
#pragma unroll
  for (int off = 16; off > 0; off >>= 1) sum += __shfl_xor(sum, off, 32);
  if (lane == 0) partial[s] = sum * coef;
}

// ---------------- Deterministic fixed-order final reduction ----------------
__global__ __launch_bounds__(256) void mmd_reduce_kernel(
    const float* __restrict__ partial, int n, float scale, float* __restrict__ out) {
  __shared__ float smem[256];
  float s = 0.0f;
  for (int i = threadIdx.x; i < n; i += 256) s += partial[i];
  smem[threadIdx.x] = s;
  __syncthreads();
  for (int stride = 128; stride > 0; stride >>= 1) {
    if ((int)threadIdx.x < stride) smem[threadIdx.x] += smem[threadIdx.x + stride];
    __syncthreads();
  }
  if (threadIdx.x == 0) out[0] = smem[0] * scale;
}

extern "C" void kernel_launch(void* const* d_in, const int* in_sizes, int n_in,
                              void* d_out, int out_size, void* d_ws, size_t ws_size,
                              hipStream_t stream) {
  const float* x = (const float*)d_in[0];
  const float* w = (const float*)d_in[1];
  const int N      = in_sizes[0] / 3;   // 8192 points
  const int nTiles = N / 16;            // 512 tiles
  const int nHalf  = nTiles / 2;        // 256 waves per symmetric Gram

  float* partial = (float*)d_ws;        // nHalf + nTiles + nHalf floats (4 KB)

  const int wpb = 8;                    // 256 threads = 8 wave32s
  dim3 block(256);
  dim3 gridSym((nHalf + wpb - 1) / wpb);
  dim3 gridFull((nTiles + wpb - 1) / wpb);

  // t1 - t2 + t3 = mean k(w,w) - 2 mean k(x,w) + mean k(x,x)
  mmd_gram_sym_kernel <<<gridSym,  block, 0, stream>>>(w,    nTiles,  1.0f, partial);
  mmd_gram_full_kernel<<<gridFull, block, 0, stream>>>(x, w, nTiles, -2.0f, partial + nHalf);
  mmd_gram_sym_kernel <<<gridSym,  block, 0, stream>>>(x,    nTiles,  1.0f, partial + nHalf + nTiles);

  const float invN2 = 1.0f / ((float)N * (float)N);
  mmd_reduce_kernel<<<1, 256, 0, stream>>>(partial, nTiles + 2 * nHalf, invN2, (float*)d_out);
}